// CovBatch_1dFV_11699490914565
// MI455X (gfx1250) — compile-verified
//
#include <hip/hip_runtime.h>
#include <hip/hip_bf16.h>

// CDNA5 / gfx1250. Reference collapses to per-row centered moments:
//   loss_b = (s2^2 - s4) / (N-1)^3,  s2 = S2 - N m^2,
//   s4 = S4 - 4 m S3 + 6 m^2 S2 - 3 N m^4,  m = S1/N,  N = V*F = 1024.
// Single pass over 512 KB of fp32; reductions ride the f32 WMMA pipe
// (V_WMMA_F32_16X16X4_F32 with B = ones acts as a 64-wide fused reducer).

typedef __attribute__((ext_vector_type(2))) float v2f;
typedef __attribute__((ext_vector_type(8))) float v8f;

#define CB_V 2
#define CB_B 128
#define CB_F 512
#define CB_N (CB_V * CB_F)   // 1024

__global__ __launch_bounds__(32)
void cov_row_loss_kernel(const float* __restrict__ zs, float* __restrict__ row_loss)
{
    const int b    = blockIdx.x;       // one wave32 per row
    const int lane = threadIdx.x;      // 0..31, full wave, EXEC all ones
    const int m    = lane & 15;        // WMMA A-matrix row owned by this lane
    const int k0   = (lane >> 4) << 1; // lanes 0-15: K=0..1, lanes 16-31: K=2..3

    v8f c1 = {}; v8f c2 = {}; v8f c3 = {}; v8f c4 = {};
    v2f ones; ones.x = 1.0f; ones.y = 1.0f;   // B matrix = all ones (4x16)

    // Row b of the logical (B, N) matrix is a 16x64 tile; K advances 4/step.
    // Logical element e maps to zs[(v*CB_B + b)*CB_F + (e % CB_F)], v = e / CB_F.
#pragma unroll
    for (int t = 0; t < CB_N / 64; ++t) {              // 16 steps
        const int e = m * 64 + t * 4 + k0;             // element index in [0, 1024)
        const int v = e >> 9;                          // which V slab (F = 512)
        const float* p = zs + ((size_t)v * CB_B + b) * CB_F + (e & (CB_F - 1));
        v2f z  = *(const v2f*)p;                       // 8B aligned: e0 is even
        v2f z2 = z * z;
        v2f z3 = z2 * z;
        v2f z4 = z2 * z2;
        // D = A x ones + C  ->  per-row running sums in fp32
        c1 = __builtin_amdgcn_wmma_f32_16x16x4_f32(false, z,  false, ones, (short)0, c1, false, false);
        c2 = __builtin_amdgcn_wmma_f32_16x16x4_f32(false, z2, false, ones, (short)0, c2, false, false);
        c3 = __builtin_amdgcn_wmma_f32_16x16x4_f32(false, z3, false, ones, (short)0, c3, false, false);
        c4 = __builtin_amdgcn_wmma_f32_16x16x4_f32(false, z4, false, ones, (short)0, c4, false, false);
    }

    // D columns are replicated; lane holds rows {r, r+8} split across halves.
    float p1 = 0.f, p2 = 0.f, p3 = 0.f, p4 = 0.f;
#pragma unroll
    for (int r = 0; r < 8; ++r) { p1 += c1[r]; p2 += c2[r]; p3 += c3[r]; p4 += c4[r]; }
    // Fold the two half-wave row groups (m=0..7 vs m=8..15).
    p1 += __shfl_xor(p1, 16, 32);
    p2 += __shfl_xor(p2, 16, 32);
    p3 += __shfl_xor(p3, 16, 32);
    p4 += __shfl_xor(p4, 16, 32);

    const float Nf    = (float)CB_N;
    const float mean  = p1 / Nf;
    const float m2    = mean * mean;
    const float s2    = p2 - Nf * m2;
    const float s4    = p4 - 4.0f * mean * p3 + 6.0f * m2 * p2 - 3.0f * Nf * m2 * m2;
    const float denom = (float)(CB_N - 1);
    const float loss  = (s2 * s2 - s4) / (denom * denom * denom);

    if (lane == 0) row_loss[b] = loss;
}

__global__ __launch_bounds__(CB_B)
void cov_mean_kernel(const float* __restrict__ row_loss, float* __restrict__ out)
{
    __shared__ float sm[CB_B];
    const int t = threadIdx.x;
    sm[t] = row_loss[t];
    __syncthreads();
#pragma unroll
    for (int s = CB_B / 2; s > 0; s >>= 1) {
        if (t < s) sm[t] += sm[t + s];
        __syncthreads();
    }
    if (t == 0) out[0] = sm[0] / (float)CB_B;
}

extern "C" void kernel_launch(void* const* d_in, const int* in_sizes, int n_in,
                              void* d_out, int out_size, void* d_ws, size_t ws_size,
                              hipStream_t stream)
{
    (void)in_sizes; (void)n_in; (void)out_size; (void)ws_size;
    const float* zs  = (const float*)d_in[0];
    float*       out = (float*)d_out;
    float*       ws  = (float*)d_ws;   // 128 per-row losses

    cov_row_loss_kernel<<<CB_B, 32, 0, stream>>>(zs, ws);
    cov_mean_kernel<<<1, CB_B, 0, stream>>>(ws, out);
}